// PosEncodingHiNeRVLocal_86036784874053
// MI455X (gfx1250) — compile-verified
//
#include <hip/hip_runtime.h>

typedef __attribute__((ext_vector_type(2))) float v2f;
typedef __attribute__((ext_vector_type(4))) float f4;
typedef __attribute__((ext_vector_type(8))) float v8f;

// Static problem constants (fixed by setup_inputs in the reference)
#define PN 2
#define PT 60
#define PH 92
#define PW 162
#define PC 64
#define KP 4
#define FANIN 224
#define MROWS (PN * PT * KP)   // 480
#define SIZE_T0 60
#define SIZE_H 270
#define SIZE_W 480
#define PATCH_T 60
#define PATCH_H 90
#define PATCH_W 160

// ---------------------------------------------------------------------------
// Kernel 1: multi-level grid lerp -> 480x224 features -> 224->64 linear,
// computed with fp32 WMMA (V_WMMA_F32_16X16X4_F32). One 128-thread workgroup
// (4 waves) per 16-row M tile; each wave owns a 16-column N tile.
// ---------------------------------------------------------------------------
__global__ void __launch_bounds__(128)
enc_wmma_kernel(const float* __restrict__ g0, const float* __restrict__ g1,
                const float* __restrict__ g2, const float* __restrict__ lwm,
                const float* __restrict__ lb, const int* __restrict__ idx,
                float* __restrict__ enc_out) {
    // +2 pad keeps row stride (226 floats = 904B) off a 64-bank multiple and
    // 8B-aligned so the paired A-frag reads become ds_load_b64.
    __shared__ float feat[16][FANIN + 2];
    const int tid = threadIdx.x;
    const int mtile = blockIdx.x;

    // Phase A: build the 16x224 feature tile in LDS (gather + 1-D lerp)
    for (int i = tid; i < 16 * FANIN; i += 128) {
        int r = i / FANIN, f = i % FANIN;
        int row = mtile * 16 + r;
        int n = row / (PT * KP);
        int rem = row % (PT * KP);
        int t = rem / KP;
        int kcell = rem % KP;
        int px_t = idx[n * 3 + 0] * PATCH_T + t;            // padding[0] == 0
        float coor = (2.0f * (float)px_t + 1.0f) / (float)SIZE_T0 - 1.0f;
        const float* g; int Tg, stride, Cg, j;
        if (f < 32)      { g = g0; Tg = 60; Cg = 32;  stride = KP * 32;  j = f;      }
        else if (f < 96) { g = g1; Tg = 30; Cg = 64;  stride = KP * 64;  j = f - 32; }
        else             { g = g2; Tg = 15; Cg = 128; stride = KP * 128; j = f - 96; }
        float pos = (coor + 1.0f) * 0.5f * (float)(Tg - 1); // align_corners=True
        int i0 = (int)floorf(pos);
        i0 = i0 < 0 ? 0 : (i0 > Tg - 1 ? Tg - 1 : i0);
        int i1 = (i0 + 1 > Tg - 1) ? Tg - 1 : i0 + 1;
        float w = pos - (float)i0;
        int col = kcell * Cg + j;
        feat[r][f] = (1.0f - w) * g[i0 * stride + col] + w * g[i1 * stride + col];
    }
    __syncthreads();

    // Phase B: D(16x16,f32) += A(16x4,f32) x B(4x16,f32), K = 224 in 56 steps.
    // fp32 WMMA operand layout: lane = {m/n = lane&15}, k = 2*(lane>>4)+vgpr.
    const int wave = tid >> 5;      // N tile 0..3
    const int lane = tid & 31;
    const int half = lane >> 4;     // 0 -> K {k,k+1}; 1 -> K {k+2,k+3}
    const int mr   = lane & 15;     // A row / B column within the tile
    const int c    = wave * 16 + mr;
    v8f acc = {};
    for (int kk = 0; kk < FANIN; kk += 4) {
        int ka = kk + 2 * half;
        v2f a = { feat[mr][ka], feat[mr][ka + 1] };
        v2f b = { lwm[c * FANIN + ka], lwm[c * FANIN + ka + 1] };
        acc = __builtin_amdgcn_wmma_f32_16x16x4_f32(false, a, false, b,
                                                    (short)0, acc, false, false);
    }
    // C/D layout: VGPR v holds M = v + 8*(lane>>4), N = lane&15
    float bias = lb[c];
    int mbase = mtile * 16 + 8 * half;
#pragma unroll
    for (int v = 0; v < 8; ++v)
        enc_out[(mbase + v) * PC + c] = acc[v] + bias;
}

// ---------------------------------------------------------------------------
// Kernel 2: out = x + mask * enc[n,t,lh*2+lw,:]  — pure HBM stream (~916 MB),
// b128 coalesced, non-temporal (stream never re-read; keep L2 clean).
// One block per (n,t,h) row; thread = (w lane, c float4 lane).
// ---------------------------------------------------------------------------
__global__ void __launch_bounds__(256)
add_local_kernel(const float* __restrict__ x, const int* __restrict__ idx,
                 const float* __restrict__ enc, float* __restrict__ out) {
    const int h = blockIdx.x, t = blockIdx.y, n = blockIdx.z;
    const int tid = threadIdx.x;
    const int c4 = tid & 15;        // 16 float4 span C = 64
    const int wi = tid >> 4;        // 16 w positions per pass

    const int idx0 = idx[n * 3 + 0];
    const int idx1 = idx[n * 3 + 1];
    const int idx2 = idx[n * 3 + 2];
    const int px_t = idx0 * PATCH_T + t;        // padding 0
    const int px_h = idx1 * PATCH_H + h - 1;    // padding 1
    const bool mt = (px_t >= 0) & (px_t < SIZE_T0);
    const bool mh = (px_h >= 0) & (px_h < SIZE_H);
    const int  lh = px_h & 1;                   // Python-style mod 2
    const bool mth = mt & mh;

    const f4* enc4 = (const f4*)enc;
    const long ebase = ((long)((n * PT + t) * KP + lh * 2) * PC) >> 2;
    const f4 e0 = enc4[ebase + c4];             // lw = 0 cell
    const f4 e1 = enc4[ebase + (PC >> 2) + c4]; // lw = 1 cell

    const f4* x4 = (const f4*)x;
    f4* o4 = (f4*)out;
    const long row4 = (long)((n * PT + t) * PH + h) * (long)(PW * PC / 4);

    for (int w = wi; w < PW; w += 16) {
        int px_w = idx2 * PATCH_W + w - 1;
        bool m = mth & (px_w >= 0) & (px_w < SIZE_W);
        f4 e = (px_w & 1) ? e1 : e0;
        long o = row4 + (long)(w * (PC >> 2) + c4);
        f4 xv = __builtin_nontemporal_load(&x4[o]);
        f4 r = m ? (xv + e) : xv;
        __builtin_nontemporal_store(r, &o4[o]);
    }
}

extern "C" void kernel_launch(void* const* d_in, const int* in_sizes, int n_in,
                              void* d_out, int out_size, void* d_ws, size_t ws_size,
                              hipStream_t stream) {
    const float* x   = (const float*)d_in[0];
    const float* g0  = (const float*)d_in[1];
    const float* g1  = (const float*)d_in[2];
    const float* g2  = (const float*)d_in[3];
    const float* lwm = (const float*)d_in[4];
    const float* lb  = (const float*)d_in[5];
    const int*   idx = (const int*)d_in[6];
    float* enc = (float*)d_ws;      // 480*64 floats = 120 KB scratch
    float* out = (float*)d_out;

    enc_wmma_kernel<<<MROWS / 16, 128, 0, stream>>>(g0, g1, g2, lwm, lb, idx, enc);
    add_local_kernel<<<dim3(PH, PT, PN), 256, 0, stream>>>(x, idx, enc, out);
}